// SINKSelfAttention_84198538870970
// MI455X (gfx1250) — compile-verified
//
#include <hip/hip_runtime.h>
#include <math.h>

// ---------------------------------------------------------------------------
// Sinkhorn attention for MI455X (gfx1250, wave32).
// B=2 H=12 S=1024 D=64, fp32. d_out = [output (B,H,S,D)] ++ [attn (B,H,S,S)].
// GEMMs via V_WMMA_F32_16X16X4_F32; Sinkhorn = streaming log-sum-exp passes.
// ---------------------------------------------------------------------------

typedef float v2f __attribute__((ext_vector_type(2)));
typedef float v8f __attribute__((ext_vector_type(8)));

#define BH    24
#define SEQ   1024
#define HD    64
#define SCALE 0.125f                 // 1/sqrt(64)
#define LOGMU (-6.93147180559945f)   // -ln(1024)  (== log_nu)

// -------------------------------- zero v -----------------------------------
__global__ void sink_zero_kernel(float* __restrict__ vv, int n) {
    int i = blockIdx.x * 256 + threadIdx.x;
    if (i < n) vv[i] = 0.0f;
}

// ------------------------- scores = scale * Q K^T (masked) -----------------
// grid = (BH*64, 8), block = (32, 8). One wave -> one 16x16 score tile.
__global__ void sink_qk_kernel(const float* __restrict__ q,
                               const float* __restrict__ k,
                               const unsigned char* __restrict__ mask,
                               float* __restrict__ attn) {
    const int bh    = blockIdx.x >> 6;
    const int mt    = blockIdx.x & 63;
    const int nt    = blockIdx.y * 8 + threadIdx.y;       // 0..63
    const int lane  = threadIdx.x;
    const int lhalf = lane >> 4;                          // 0 or 1
    const int lmod  = lane & 15;

    const float* qb = q + ((size_t)bh * SEQ + mt * 16 + lmod) * HD;
    const float* kb = k + ((size_t)bh * SEQ + nt * 16 + lmod) * HD;

    v8f c = {};
#pragma unroll
    for (int kk = 0; kk < HD; kk += 4) {
        v2f a, b;
        a.x = qb[kk + 2 * lhalf + 0];
        a.y = qb[kk + 2 * lhalf + 1];
        b.x = kb[kk + 2 * lhalf + 0];
        b.y = kb[kk + 2 * lhalf + 1];
        c = __builtin_amdgcn_wmma_f32_16x16x4_f32(false, a, false, b,
                                                  (short)0, c, false, false);
    }

    float* out = attn + (size_t)bh * SEQ * SEQ;
    const int n = nt * 16 + lmod;
#pragma unroll
    for (int r = 0; r < 8; ++r) {
        const int m = mt * 16 + 8 * lhalf + r;
        float val = c[r] * SCALE;
        if (mask[(size_t)m * SEQ + n]) val = -1e9f;
        out[(size_t)m * SEQ + n] = val;
    }
}

// --------------------- u[i] = logmu - LSE_j(Sc[i][j] + v[j]) ---------------
// grid = BH*SEQ blocks, 256 threads: one block per (head,row).
__global__ void sink_row_lse_kernel(const float* __restrict__ attn,
                                    const float* __restrict__ vv,
                                    float* __restrict__ uu) {
    const int bh  = blockIdx.x >> 10;
    const int row = blockIdx.x & 1023;
    const float* base = attn + (size_t)bh * SEQ * SEQ + (size_t)row * SEQ;
    const float* vb   = vv + bh * SEQ;

    float m = -INFINITY, s = 0.0f;
#pragma unroll
    for (int j = threadIdx.x; j < SEQ; j += 256) {
        float x = base[j] + vb[j];
        if (x > m) { s = s * expf(m - x) + 1.0f; m = x; }
        else       { s += expf(x - m); }
    }

    __shared__ float sm[256];
    __shared__ float ss[256];
    sm[threadIdx.x] = m; ss[threadIdx.x] = s;
    __syncthreads();
    for (int off = 128; off > 0; off >>= 1) {
        if (threadIdx.x < off) {
            float m1 = sm[threadIdx.x], s1 = ss[threadIdx.x];
            float m2 = sm[threadIdx.x + off], s2 = ss[threadIdx.x + off];
            float M = fmaxf(m1, m2);
            ss[threadIdx.x] = s1 * expf(m1 - M) + s2 * expf(m2 - M);
            sm[threadIdx.x] = M;
        }
        __syncthreads();
    }
    if (threadIdx.x == 0)
        uu[bh * SEQ + row] = LOGMU - (sm[0] + logf(ss[0]));
}

// --------------------- v[j] = lognu - LSE_i(Sc[i][j] + u[i]) ---------------
// grid = BH*4 blocks, 256 threads: thread owns one column, coalesced rows.
__global__ void sink_col_lse_kernel(const float* __restrict__ attn,
                                    const float* __restrict__ uu,
                                    float* __restrict__ vv) {
    const int bh = blockIdx.x >> 2;
    const int j  = (blockIdx.x & 3) * 256 + threadIdx.x;
    const float* base = attn + (size_t)bh * SEQ * SEQ + j;
    const float* ub   = uu + bh * SEQ;

    float m = -INFINITY, s = 0.0f;
#pragma unroll 4
    for (int i = 0; i < SEQ; ++i) {
        float x = base[(size_t)i * SEQ] + ub[i];
        if (x > m) { s = s * expf(m - x) + 1.0f; m = x; }
        else       { s += expf(x - m); }
    }
    vv[bh * SEQ + j] = LOGMU - (m + logf(s));
}

// --------------- attn = exp(Sc + u_i + v_j) * S  (in place) ----------------
// One float4 per thread; 1024 % 4 == 0 so a float4 stays inside one row.
__global__ void sink_finalize_kernel(float* __restrict__ attn,
                                     const float* __restrict__ uu,
                                     const float* __restrict__ vv) {
    const size_t idx = ((size_t)blockIdx.x * 256 + threadIdx.x) * 4;
    const int bh = (int)(idx >> 20);
    const int rem = (int)(idx & 1048575);
    const int i = rem >> 10;
    const int j = rem & 1023;
    const float u = uu[bh * SEQ + i];
    const float4 v4 = *(const float4*)(vv + bh * SEQ + j);
    float4 sc = *(float4*)(attn + idx);
    sc.x = expf(sc.x + u + v4.x) * (float)SEQ;
    sc.y = expf(sc.y + u + v4.y) * (float)SEQ;
    sc.z = expf(sc.z + u + v4.z) * (float)SEQ;
    sc.w = expf(sc.w + u + v4.w) * (float)SEQ;
    *(float4*)(attn + idx) = sc;
}

// ------------------------------ out = attn @ V -----------------------------
// grid = BH*16 blocks, block = (32, 4). One wave -> a full 16x64 output strip
// (4 accumulators). attn fragment (A) loaded ONCE per k-step and shared by
// 4 WMMAs -> attn matrix streamed exactly once (96 MB, roofline minimum),
// and 4 independent WMMA chains per wave hide the accumulate latency.
__global__ void sink_av_kernel(const float* __restrict__ attn,
                               const float* __restrict__ v,
                               float* __restrict__ out) {
    const int bh    = blockIdx.x >> 4;
    const int mt    = (blockIdx.x & 15) * 4 + threadIdx.y;  // 0..63
    const int lane  = threadIdx.x;
    const int lhalf = lane >> 4;
    const int lmod  = lane & 15;

    const float* ab = attn + (size_t)bh * SEQ * SEQ + (size_t)(mt * 16 + lmod) * SEQ;
    const float* vb = v + (size_t)bh * SEQ * HD + lmod;     // + nt*16 added below

    v8f c0 = {}, c1 = {}, c2 = {}, c3 = {};
#pragma unroll 4
    for (int kk = 0; kk < SEQ; kk += 4) {
        v2f a;
        a.x = ab[kk + 2 * lhalf + 0];
        a.y = ab[kk + 2 * lhalf + 1];
        const float* vrow = vb + (size_t)(kk + 2 * lhalf) * HD;  // row k, row k+1 at +HD
        v2f b0, b1, b2, b3;
        b0.x = vrow[0];       b0.y = vrow[HD + 0];
        b1.x = vrow[16];      b1.y = vrow[HD + 16];
        b2.x = vrow[32];      b2.y = vrow[HD + 32];
        b3.x = vrow[48];      b3.y = vrow[HD + 48];
        c0 = __builtin_amdgcn_wmma_f32_16x16x4_f32(false, a, false, b0, (short)0, c0, false, false);
        c1 = __builtin_amdgcn_wmma_f32_16x16x4_f32(false, a, false, b1, (short)0, c1, false, false);
        c2 = __builtin_amdgcn_wmma_f32_16x16x4_f32(false, a, false, b2, (short)0, c2, false, false);
        c3 = __builtin_amdgcn_wmma_f32_16x16x4_f32(false, a, false, b3, (short)0, c3, false, false);
    }

    float* ob = out + (size_t)bh * SEQ * HD + lmod;
#pragma unroll
    for (int r = 0; r < 8; ++r) {
        const size_t mrow = (size_t)(mt * 16 + 8 * lhalf + r) * HD;
        ob[mrow +  0] = c0[r];
        ob[mrow + 16] = c1[r];
        ob[mrow + 32] = c2[r];
        ob[mrow + 48] = c3[r];
    }
}

// ---------------------------------------------------------------------------
extern "C" void kernel_launch(void* const* d_in, const int* in_sizes, int n_in,
                              void* d_out, int out_size, void* d_ws, size_t ws_size,
                              hipStream_t stream) {
    const float* q = (const float*)d_in[0];
    const float* k = (const float*)d_in[1];
    const float* v = (const float*)d_in[2];
    const unsigned char* mask = (const unsigned char*)d_in[3];

    float* out_o = (float*)d_out;                       // (B,H,S,D)
    float* attn  = out_o + (size_t)BH * SEQ * HD;       // (B,H,S,S)

    float* uu = (float*)d_ws;                           // BH*SEQ floats
    float* vv = uu + BH * SEQ;                          // BH*SEQ floats

    // v = 0
    sink_zero_kernel<<<(BH * SEQ + 255) / 256, 256, 0, stream>>>(vv, BH * SEQ);

    // scores (masked, scaled) -> attn region
    sink_qk_kernel<<<dim3(BH * 64, 8), dim3(32, 8), 0, stream>>>(q, k, mask, attn);

    // 3 Sinkhorn iterations (u-update then v-update; u cancels analytically)
    for (int it = 0; it < 3; ++it) {
        sink_row_lse_kernel<<<BH * SEQ, 256, 0, stream>>>(attn, vv, uu);
        sink_col_lse_kernel<<<BH * 4, 256, 0, stream>>>(attn, uu, vv);
    }

    // pi * S  (in place)
    sink_finalize_kernel<<<(BH * SEQ * SEQ / 4 + 255) / 256, 256, 0, stream>>>(attn, uu, vv);

    // output = attn @ V  (one wave per 16-row strip, attn streamed once)
    sink_av_kernel<<<BH * 16, dim3(32, 4), 0, stream>>>(attn, v, out_o);
}